// SubgraphPooling_80633716015124
// MI455X (gfx1250) — compile-verified
//
#include <hip/hip_runtime.h>

// Segment-mean (gather + scatter-mean over sorted segment ids) for MI455X.
// Sorted segment ids => rows for 16 consecutive segments form one contiguous
// window of the gather stream; the segmented sum is done as
//   C(16 segs x 16 dims) += A(16x4 one-hot selector) x B(4 rows x 16 dims)
// with V_WMMA_F32_16X16X4_F32 (f32 end-to-end, matches f32 reference).
// Each wave: one 16-segment tile x one 32-dim slice (2 accumulator tiles),
// 2 chunks (8 rows) per loop iteration for memory-level parallelism.

typedef __attribute__((ext_vector_type(2))) float v2f;
typedef __attribute__((ext_vector_type(8))) float v8f;

#define D_FEAT 128

__device__ __forceinline__ int imin(int a, int b) { return a < b ? a : b; }

__device__ __forceinline__ int lower_bound_i32(const int* __restrict__ a, int n, int v) {
  int lo = 0, hi = n;
  while (lo < hi) {
    int mid = (lo + hi) >> 1;
    if (a[mid] < v) lo = mid + 1; else hi = mid;
  }
  return lo;
}

__global__ __launch_bounds__(256) void SubgraphPooling_seg_mean_wmma(
    const float* __restrict__ feat,   // [n_nodes, 128]
    const int*   __restrict__ ids,    // [n_gather] gather row ids
    const int*   __restrict__ seg,    // [n_gather] sorted segment ids
    float*       __restrict__ out,    // [n_macro, 128]
    int n_gather, int n_macro)
{
  const int lane  = threadIdx.x & 31;
  const int wave  = threadIdx.x >> 5;
  const int gwave = blockIdx.x * (blockDim.x >> 5) + wave;

  const int seg_tile = gwave >> 2;   // 16 segments per tile
  const int dim_pair = gwave & 3;    // which 32-dim slice (2 x 16-dim tiles)
  const int m0 = seg_tile * 16;
  if (m0 >= n_macro) return;         // wave-uniform exit (EXEC all-ones for WMMA)

  // --- segment boundaries: lane l in [0,16] finds lower_bound(seg, m0+l) ---
  const int tl = (lane < 16) ? lane : 16;
  const int sB = lower_bound_i32(seg, n_gather, m0 + tl);
  const int start = __builtin_amdgcn_readlane(sB, 0);    // scalar (SGPR) bounds
  const int end   = __builtin_amdgcn_readlane(sB, 16);
  const int s_next = __shfl(sB, (lane & 15) + 1, 32);
  const int cnt    = s_next - sB;                        // valid on lanes 0..15
  const float inv  = (cnt > 0) ? 1.0f / (float)cnt : 0.0f;

  // --- WMMA lane mapping ---
  const int hi = lane >> 4;          // 0: K=0/1, M-block 0..7 ; 1: K=2/3, M-block 8..15
  const int nl = lane & 15;          // A: M row; B/C: N column
  const int mtag = m0 + nl;          // segment id this lane selects for in A
  const int t0   = dim_pair * 2;
  const int col0 = t0 * 16 + nl;
  const int col1 = col0 + 16;

  v8f acc0 = {};                     // dims [t0*16, t0*16+16)
  v8f acc1 = {};                     // dims [t0*16+16, t0*16+32)

  const int last = n_gather - 1;
  for (int k0 = start; k0 < end; k0 += 8) {              // 8 gather rows / iter
    v2f a[2], b0[2], b1[2];
#pragma unroll
    for (int c = 0; c < 2; ++c) {                        // build 2 chunks of 4 rows
      const int ka  = k0 + c * 4 + hi * 2;               // this lane's K rows
      const int ka0 = imin(ka, last);                    // branchless tail clamp
      const int ka1 = imin(ka + 1, last);

      const int sg0 = seg[ka0];                          // unconditional loads
      const int sg1 = seg[ka1];
      const int id0 = ids[ka0];
      const int id1 = ids[ka1];

      // branchless one-hot selector (bitwise & avoids short-circuit branches)
      const int in0 = (ka     < end) ? 1 : 0;
      const int in1 = (ka + 1 < end) ? 1 : 0;
      a[c].x = (in0 & (int)(sg0 == mtag)) ? 1.0f : 0.0f;
      a[c].y = (in1 & (int)(sg1 == mtag)) ? 1.0f : 0.0f;

      const float* r0 = feat + (size_t)id0 * D_FEAT;
      const float* r1 = feat + (size_t)id1 * D_FEAT;
      b0[c].x = r0[col0]; b0[c].y = r1[col0];
      b1[c].x = r0[col1]; b1[c].y = r1[col1];
    }
#pragma unroll
    for (int c = 0; c < 2; ++c) {
      acc0 = __builtin_amdgcn_wmma_f32_16x16x4_f32(false, a[c], false, b0[c],
                                                   (short)0, acc0, false, false);
      acc1 = __builtin_amdgcn_wmma_f32_16x16x4_f32(false, a[c], false, b1[c],
                                                   (short)0, acc1, false, false);
    }
  }

  // --- scale by 1/count and store: C layout M = v + 8*hi, N = nl ---
#pragma unroll
  for (int v = 0; v < 8; ++v) {
    const int   M   = v + hi * 8;
    const float iv  = __shfl(inv, M, 32);   // counts live on lanes 0..15
    const int   row = m0 + M;
    if (row < n_macro) {
      out[(size_t)row * D_FEAT + col0] = acc0[v] * iv;
      out[(size_t)row * D_FEAT + col1] = acc1[v] * iv;
    }
  }
}

extern "C" void kernel_launch(void* const* d_in, const int* in_sizes, int n_in,
                              void* d_out, int out_size, void* d_ws, size_t ws_size,
                              hipStream_t stream) {
  const float* feat = (const float*)d_in[0];
  const int*   ids  = (const int*)d_in[1];
  const int*   seg  = (const int*)d_in[2];
  float*       out  = (float*)d_out;

  const int n_gather = in_sizes[1];
  const int n_macro  = out_size / D_FEAT;

  const int n_tiles = (n_macro + 15) / 16;
  const int n_waves = n_tiles * 4;                 // 4 dim-slices per tile
  const int waves_per_block = 8;                   // 256 threads, 8 wave32s
  const int blocks = (n_waves + waves_per_block - 1) / waves_per_block;

  SubgraphPooling_seg_mean_wmma<<<blocks, waves_per_block * 32, 0, stream>>>(
      feat, ids, seg, out, n_gather, n_macro);
}